// DipolePredictorSE3_73349451481425
// MI455X (gfx1250) — compile-verified
//
#include <hip/hip_runtime.h>

#define N_NODES 100000
#define N_EDGES 6400000
#define HIDDEN 16
#define HEADS 8
#define DEPTH 2
#define NTILES (N_NODES / 16)   // 6250, exact

typedef __attribute__((ext_vector_type(2))) float v2f;
typedef __attribute__((ext_vector_type(4))) float v4f;
typedef __attribute__((ext_vector_type(8))) float v8f;

// ---- order-preserving float <-> uint key for atomicMax segment-max ----
__device__ __forceinline__ unsigned f2key(float f) {
    unsigned u = __float_as_uint(f);
    return (u & 0x80000000u) ? ~u : (u | 0x80000000u);
}
__device__ __forceinline__ float key2f(unsigned kk) {
    unsigned u = (kk & 0x80000000u) ? (kk & 0x7FFFFFFFu) : ~kk;
    return __uint_as_float(u);
}

// ---- h = x @ W_in : [N,4] @ [4,16] ----
__global__ void k_input_proj(const float* __restrict__ x, const float* __restrict__ Win,
                             float* __restrict__ h) {
    int n = blockIdx.x * blockDim.x + threadIdx.x;
    if (n >= N_NODES) return;
    float4 xr = *(const float4*)(x + n * 4);
    float xv[4] = {xr.x, xr.y, xr.z, xr.w};
#pragma unroll
    for (int j = 0; j < 16; ++j) {
        float acc = 0.f;
#pragma unroll
        for (int d = 0; d < 4; ++d) acc += xv[d] * Win[d * 16 + j];
        h[n * 16 + j] = acc;
    }
}

// ---- q,k,v = h @ Wq/Wk/Wv via chained V_WMMA_F32_16X16X4_F32 ----
// One wave handles one 16-row node tile; K=16 -> 4 chained K=4 WMMAs per matrix.
__global__ void __launch_bounds__(128)
k_qkv(const float* __restrict__ h,
      const float* __restrict__ Wq, const float* __restrict__ Wk, const float* __restrict__ Wv,
      float* __restrict__ q, float* __restrict__ k, float* __restrict__ v) {
    int wave = threadIdx.x >> 5;
    int tile = blockIdx.x * 4 + wave;
    if (tile >= NTILES) return;               // wave-uniform: EXEC stays all-ones
    int lane = threadIdx.x & 31;
    int c    = lane & 15;                     // A: row M ; B: col N ; D: col N
    int hi   = lane >> 4;                     // lane half selects K sub-pair / M+8
    int base = tile * 16;

    v8f aq = {}, ak = {}, av = {};
#pragma unroll
    for (int kk = 0; kk < 4; ++kk) {
        int kb = 4 * kk + 2 * hi;
        float2 ha = *(const float2*)(h + (base + c) * 16 + kb);
        v2f A;  A.x = ha.x;            A.y = ha.y;
        v2f Bq; Bq.x = Wq[kb * 16 + c]; Bq.y = Wq[(kb + 1) * 16 + c];
        v2f Bk; Bk.x = Wk[kb * 16 + c]; Bk.y = Wk[(kb + 1) * 16 + c];
        v2f Bv; Bv.x = Wv[kb * 16 + c]; Bv.y = Wv[(kb + 1) * 16 + c];
        aq = __builtin_amdgcn_wmma_f32_16x16x4_f32(false, A, false, Bq, (short)0, aq, false, false);
        ak = __builtin_amdgcn_wmma_f32_16x16x4_f32(false, A, false, Bk, (short)0, ak, false, false);
        av = __builtin_amdgcn_wmma_f32_16x16x4_f32(false, A, false, Bv, (short)0, av, false, false);
    }
#pragma unroll
    for (int r = 0; r < 8; ++r) {
        int m = base + r + 8 * hi;
        q[m * 16 + c] = aq[r];
        k[m * 16 + c] = ak[r];
        v[m * 16 + c] = av[r];
    }
}

__device__ __forceinline__ void load_row16(const float* __restrict__ p, float* r) {
    const float4* p4 = (const float4*)p;
#pragma unroll
    for (int i = 0; i < 4; ++i) {
        float4 t = p4[i];
        r[4 * i] = t.x; r[4 * i + 1] = t.y; r[4 * i + 2] = t.z; r[4 * i + 3] = t.w;
    }
}

// ---- pass A: segment max of attention scores (order-preserving uint atomicMax) ----
// Edge indices are single-use streams: load non-temporal (TH=NT) so the 32 MB of
// gathered node arrays stays resident in L2 while 51 MB/pass of indices streams by.
__global__ void k_edge_smax(const int* __restrict__ ei, const float* __restrict__ q,
                            const float* __restrict__ k, unsigned* __restrict__ smaxk) {
    long long e = (long long)blockIdx.x * blockDim.x + threadIdx.x;
    if (e >= N_EDGES) return;
    int s = __builtin_nontemporal_load(ei + e);
    int d = __builtin_nontemporal_load(ei + N_EDGES + e);
    float qr[16], kr[16];
    load_row16(q + d * 16, qr);
    load_row16(k + s * 16, kr);
    const float scale = 0.70710678118f;       // 1/sqrt(HEAD_DIM=2)
#pragma unroll
    for (int hh = 0; hh < HEADS; ++hh) {
        float sc = (qr[2 * hh] * kr[2 * hh] + qr[2 * hh + 1] * kr[2 * hh + 1]) * scale;
        atomicMax(smaxk + d * 8 + hh, f2key(sc));
    }
}

// ---- pass B: denom = segment_sum(exp(score - smax)) ----
__global__ void k_edge_expsum(const int* __restrict__ ei, const float* __restrict__ q,
                              const float* __restrict__ k, const unsigned* __restrict__ smaxk,
                              float* __restrict__ denom) {
    long long e = (long long)blockIdx.x * blockDim.x + threadIdx.x;
    if (e >= N_EDGES) return;
    int s = __builtin_nontemporal_load(ei + e);
    int d = __builtin_nontemporal_load(ei + N_EDGES + e);
    float qr[16], kr[16];
    load_row16(q + d * 16, qr);
    load_row16(k + s * 16, kr);
    const float scale = 0.70710678118f;
#pragma unroll
    for (int hh = 0; hh < HEADS; ++hh) {
        float sc = (qr[2 * hh] * kr[2 * hh] + qr[2 * hh + 1] * kr[2 * hh + 1]) * scale;
        float eh = __expf(sc - key2f(smaxk[d * 8 + hh]));
        atomicAdd(denom + d * 8 + hh, eh);
    }
}

// ---- pass C: agg = segment_sum(alpha * v[src]) ----
__global__ void k_edge_agg(const int* __restrict__ ei, const float* __restrict__ q,
                           const float* __restrict__ k, const float* __restrict__ v,
                           const unsigned* __restrict__ smaxk, const float* __restrict__ denom,
                           float* __restrict__ agg) {
    long long e = (long long)blockIdx.x * blockDim.x + threadIdx.x;
    if (e >= N_EDGES) return;
    int s = __builtin_nontemporal_load(ei + e);
    int d = __builtin_nontemporal_load(ei + N_EDGES + e);
    float qr[16], kr[16], vr[16];
    load_row16(q + d * 16, qr);
    load_row16(k + s * 16, kr);
    load_row16(v + s * 16, vr);
    const float scale = 0.70710678118f;
#pragma unroll
    for (int hh = 0; hh < HEADS; ++hh) {
        float sc = (qr[2 * hh] * kr[2 * hh] + qr[2 * hh + 1] * kr[2 * hh + 1]) * scale;
        float eh = __expf(sc - key2f(smaxk[d * 8 + hh]));
        float alpha = eh / (denom[d * 8 + hh] + 1e-9f);
        atomicAdd(agg + d * 16 + 2 * hh,     alpha * vr[2 * hh]);
        atomicAdd(agg + d * 16 + 2 * hh + 1, alpha * vr[2 * hh + 1]);
    }
}

// ---- htmp = h + agg @ Wo (WMMA) ----
__global__ void __launch_bounds__(128)
k_out_res(const float* __restrict__ h, const float* __restrict__ agg,
          const float* __restrict__ Wo, float* __restrict__ htmp) {
    int wave = threadIdx.x >> 5;
    int tile = blockIdx.x * 4 + wave;
    if (tile >= NTILES) return;
    int lane = threadIdx.x & 31;
    int c    = lane & 15;
    int hi   = lane >> 4;
    int base = tile * 16;

    v8f acc = {};
#pragma unroll
    for (int kk = 0; kk < 4; ++kk) {
        int kb = 4 * kk + 2 * hi;
        float2 aa = *(const float2*)(agg + (base + c) * 16 + kb);
        v2f A; A.x = aa.x;             A.y = aa.y;
        v2f B; B.x = Wo[kb * 16 + c];  B.y = Wo[(kb + 1) * 16 + c];
        acc = __builtin_amdgcn_wmma_f32_16x16x4_f32(false, A, false, B, (short)0, acc, false, false);
    }
#pragma unroll
    for (int r = 0; r < 8; ++r) {
        int m = base + r + 8 * hi;
        htmp[m * 16 + c] = h[m * 16 + c] + acc[r];
    }
}

// ---- h = LayerNorm(htmp) ----
__global__ void k_layernorm(const float* __restrict__ htmp, float* __restrict__ h) {
    int n = blockIdx.x * blockDim.x + threadIdx.x;
    if (n >= N_NODES) return;
    float r[16];
    load_row16(htmp + n * 16, r);
    float mu = 0.f;
#pragma unroll
    for (int j = 0; j < 16; ++j) mu += r[j];
    mu *= (1.f / 16.f);
    float var = 0.f;
#pragma unroll
    for (int j = 0; j < 16; ++j) { float t = r[j] - mu; var += t * t; }
    var *= (1.f / 16.f);
    float inv = rsqrtf(var + 1e-5f);
#pragma unroll
    for (int j = 0; j < 16; ++j) h[n * 16 + j] = (r[j] - mu) * inv;
}

// ---- pooled += sum over nodes of (h @ W_out) ----
__global__ void k_pool(const float* __restrict__ h, const float* __restrict__ Wout,
                       float* __restrict__ pooled) {
    int n = blockIdx.x * blockDim.x + threadIdx.x;
    float y[4] = {0.f, 0.f, 0.f, 0.f};
    if (n < N_NODES) {
        const v4f* h4 = (const v4f*)(h + n * 16);
#pragma unroll
        for (int i = 0; i < 4; ++i) {
            v4f t = __builtin_nontemporal_load(h4 + i);   // last use of h
            float hv[4] = {t.x, t.y, t.z, t.w};
#pragma unroll
            for (int jj = 0; jj < 4; ++jj)
#pragma unroll
                for (int d = 0; d < 4; ++d)
                    y[d] += hv[jj] * Wout[(4 * i + jj) * 4 + d];
        }
    }
#pragma unroll
    for (int off = 16; off > 0; off >>= 1)
#pragma unroll
        for (int d = 0; d < 4; ++d) y[d] += __shfl_down(y[d], off);
    __shared__ float sred[8][4];
    int lane = threadIdx.x & 31, wave = threadIdx.x >> 5;
    if (lane == 0)
        for (int d = 0; d < 4; ++d) sred[wave][d] = y[d];
    __syncthreads();
    if (threadIdx.x == 0) {
        float t[4] = {0.f, 0.f, 0.f, 0.f};
        int nw = (blockDim.x + 31) / 32;
        for (int w = 0; w < nw; ++w)
            for (int d = 0; d < 4; ++d) t[d] += sred[w][d];
        for (int d = 0; d < 4; ++d) atomicAdd(&pooled[d], t[d]);
    }
}

// ---- out[3] = (pooled/N) @ fc_w + fc_b ----
__global__ void k_head(const float* __restrict__ pooled, const float* __restrict__ fcw,
                       const float* __restrict__ fcb, float* __restrict__ out) {
    int cidx = threadIdx.x;
    if (cidx < 3) {
        float acc = fcb[cidx];
#pragma unroll
        for (int d = 0; d < 4; ++d)
            acc += (pooled[d] * (1.0f / N_NODES)) * fcw[d * 3 + cidx];
        out[cidx] = acc;
    }
}

extern "C" void kernel_launch(void* const* d_in, const int* in_sizes, int n_in,
                              void* d_out, int out_size, void* d_ws, size_t ws_size,
                              hipStream_t stream) {
    (void)in_sizes; (void)n_in; (void)out_size; (void)ws_size;
    const float* x    = (const float*)d_in[0];
    const int*   ei   = (const int*)d_in[1];
    const float* Win  = (const float*)d_in[2];
    const float* Wq   = (const float*)d_in[3];
    const float* Wk   = (const float*)d_in[4];
    const float* Wv   = (const float*)d_in[5];
    const float* Wo   = (const float*)d_in[6];
    const float* Wout = (const float*)d_in[7];
    const float* fcw  = (const float*)d_in[8];
    const float* fcb  = (const float*)d_in[9];
    float* out = (float*)d_out;

    // workspace: ~44.8 MB of fp32/uint scratch
    float*    h      = (float*)d_ws;
    float*    q      = h    + (size_t)N_NODES * 16;
    float*    k      = q    + (size_t)N_NODES * 16;
    float*    v      = k    + (size_t)N_NODES * 16;
    float*    agg    = v    + (size_t)N_NODES * 16;
    float*    htmp   = agg  + (size_t)N_NODES * 16;
    float*    denom  = htmp + (size_t)N_NODES * 16;
    unsigned* smaxk  = (unsigned*)(denom + (size_t)N_NODES * 8);
    float*    pooled = (float*)(smaxk + (size_t)N_NODES * 8);

    const int TPB = 256;
    int nodeBlocks = (N_NODES + TPB - 1) / TPB;
    int edgeBlocks = (N_EDGES + TPB - 1) / TPB;
    int tileBlocks = (NTILES + 3) / 4;

    k_input_proj<<<nodeBlocks, TPB, 0, stream>>>(x, Win, h);

    for (int l = 0; l < DEPTH; ++l) {
        (void)hipMemsetAsync(smaxk, 0, (size_t)N_NODES * 8 * sizeof(unsigned), stream);
        (void)hipMemsetAsync(denom, 0, (size_t)N_NODES * 8 * sizeof(float), stream);
        (void)hipMemsetAsync(agg,   0, (size_t)N_NODES * 16 * sizeof(float), stream);
        k_qkv<<<tileBlocks, 128, 0, stream>>>(h, Wq + l * 256, Wk + l * 256, Wv + l * 256, q, k, v);
        k_edge_smax  <<<edgeBlocks, TPB, 0, stream>>>(ei, q, k, smaxk);
        k_edge_expsum<<<edgeBlocks, TPB, 0, stream>>>(ei, q, k, smaxk, denom);
        k_edge_agg   <<<edgeBlocks, TPB, 0, stream>>>(ei, q, k, v, smaxk, denom, agg);
        k_out_res<<<tileBlocks, 128, 0, stream>>>(h, agg, Wo + l * 256, htmp);
        k_layernorm<<<nodeBlocks, TPB, 0, stream>>>(htmp, h);
    }

    (void)hipMemsetAsync(pooled, 0, 4 * sizeof(float), stream);
    k_pool<<<nodeBlocks, TPB, 0, stream>>>(h, Wout, pooled);
    k_head<<<1, 32, 0, stream>>>(pooled, fcw, fcb, out);
}